// DeformConv2d_65953517797552
// MI455X (gfx1250) — compile-verified
//
#include <hip/hip_runtime.h>
#include <math.h>

// Problem constants
#define BN    4
#define CIN   256
#define COUT  256
#define HH    64
#define WW    64
#define HW    (HH*WW)          // 4096
#define KKN   9
#define KDIM  (CIN*KKN)        // 2304
#define NOFF  18               // 2*KK offset channels

typedef __attribute__((ext_vector_type(2))) float v2f;
typedef __attribute__((ext_vector_type(8))) float v8f;

// ---------------------------------------------------------------------------
// Kernel 1: offset field = conv2d(x, w_off) + b_off   (NCHW / OIHW, pad=1)
// One thread per (b,h,w) position holding all 18 output-channel accumulators:
// each x value is loaded once and FMA'd vs 18 wave-uniform (scalar) weights.
// ---------------------------------------------------------------------------
__global__ __launch_bounds__(256)
void offset_conv_kernel(const float* __restrict__ x,
                        const float* __restrict__ w_off,
                        const float* __restrict__ b_off,
                        float* __restrict__ offs) {
  int idx = blockIdx.x * blockDim.x + threadIdx.x;   // B*64*64 = 16384
  int w = idx & 63;
  int h = (idx >> 6) & 63;
  int b = idx >> 12;
  if (b >= BN) return;

  float acc[NOFF];
  #pragma unroll
  for (int oc = 0; oc < NOFF; ++oc) acc[oc] = b_off[oc];

  const float* xb = x + (size_t)b * (CIN * HW);
  for (int c = 0; c < CIN; ++c) {
    const float* xc = xb + c * HW;
    const float* wc = w_off + c * KKN;   // + oc*CIN*KKN per channel (uniform)
    #pragma unroll
    for (int kh = 0; kh < 3; ++kh) {
      int hy = h + kh - 1;
      if (hy < 0 || hy >= HH) continue;
      const float* xr = xc + hy * WW;
      #pragma unroll
      for (int kw = 0; kw < 3; ++kw) {
        int wx = w + kw - 1;
        if (wx < 0 || wx >= WW) continue;
        float xv = xr[wx];
        int kk = kh * 3 + kw;
        #pragma unroll
        for (int oc = 0; oc < NOFF; ++oc)
          acc[oc] = fmaf(xv, wc[(size_t)oc * (CIN * KKN) + kk], acc[oc]);
      }
    }
  }
  float* ob = offs + ((size_t)b * NOFF * HH + h) * WW + w;
  #pragma unroll
  for (int oc = 0; oc < NOFF; ++oc) ob[(size_t)oc * HW] = acc[oc];
}

// ---------------------------------------------------------------------------
// Kernel 2: repack w(O,C,3,3) -> Wt2[k/2][oc][2] with k = c*9+kk.
// K-pair interleave makes each WMMA B-fragment one coalesced global_load_b64.
// ---------------------------------------------------------------------------
__global__ __launch_bounds__(256)
void repack_w_kernel(const float* __restrict__ w, float* __restrict__ Wt) {
  int idx = blockIdx.x * blockDim.x + threadIdx.x;   // 2304*256 = 589824
  if (idx >= KDIM * COUT) return;
  int oc = idx & 255;
  int k  = idx >> 8;
  int c  = k / KKN;
  int kk = k - c * KKN;
  Wt[((size_t)(k >> 1) * COUT + oc) * 2 + (k & 1)] = w[(oc * CIN + c) * KKN + kk];
}

// ---------------------------------------------------------------------------
// Kernel 3: fused bilinear sampling + GEMM via V_WMMA_F32_16X16X4_F32.
// Block = 256 threads (8 waves) = one 16-wide spatial strip (b, h, w0..w0+15),
// all 256 output channels. Wave wv owns oc tiles [32wv,32wv+16),[32wv+16,32wv+32).
// A panel staged in LDS, K-pair interleaved so fragments are ds_load_b64.
// ---------------------------------------------------------------------------
__global__ __launch_bounds__(256)
void deform_wmma_kernel(const float* __restrict__ x,
                        const float* __restrict__ offs,
                        const float* __restrict__ Wt,
                        const float* __restrict__ bias,
                        float* __restrict__ out) {
  __shared__ int   sO[4][144];     // clamped corner offsets into 64x64 image
  __shared__ float sW[4][144];     // bilinear weights (mask folded in)
  __shared__ float A[72 * 32];     // A chunk: [kpair][m][2]  (2304 floats)

  int bid = blockIdx.x;            // 1024 = 4 strips * 64 rows * 4 images
  int w0  = (bid & 3) * 16;
  int h   = (bid >> 2) & 63;
  int b   = bid >> 8;
  int tid = threadIdx.x;

  // ---- phase 0: per-(m,kk) sampling parameters (144 entries) ----
  if (tid < 144) {
    int m  = tid & 15;
    int kk = tid >> 4;             // 0..8
    int kh = kk / 3, kw = kk - kh * 3;
    int wpos = w0 + m;
    const float* ob = offs + ((b * NOFF + 2 * kk) * HH + h) * WW + wpos;
    float dy = ob[0];
    float dx = ob[HW];             // next channel (2*kk+1)
    float py = (float)(h - 1 + kh) + dy;
    float px = (float)(wpos - 1 + kw) + dx;
    float fy = floorf(py), fx = floorf(px);
    float ly = py - fy,    lx = px - fx;
    int y0 = (int)fy, x0 = (int)fx;
    int y1 = y0 + 1,  x1 = x0 + 1;
    bool vy0 = (y0 >= 0) & (y0 < HH);
    bool vy1 = (y1 >= 0) & (y1 < HH);
    bool vx0 = (x0 >= 0) & (x0 < WW);
    bool vx1 = (x1 >= 0) & (x1 < WW);
    int y0c = min(max(y0, 0), HH - 1), y1c = min(max(y1, 0), HH - 1);
    int x0c = min(max(x0, 0), WW - 1), x1c = min(max(x1, 0), WW - 1);
    sO[0][tid] = y0c * WW + x0c;
    sO[1][tid] = y0c * WW + x1c;
    sO[2][tid] = y1c * WW + x0c;
    sO[3][tid] = y1c * WW + x1c;
    float wy0 = 1.f - ly, wy1 = ly, wx0 = 1.f - lx, wx1 = lx;
    sW[0][tid] = (vy0 && vx0) ? wy0 * wx0 : 0.f;
    sW[1][tid] = (vy0 && vx1) ? wy0 * wx1 : 0.f;
    sW[2][tid] = (vy1 && vx0) ? wy1 * wx0 : 0.f;
    sW[3][tid] = (vy1 && vx1) ? wy1 * wx1 : 0.f;
  }
  __syncthreads();

  int lane  = tid & 31;
  int wv    = tid >> 5;
  int mlane = lane & 15;
  int klo   = (lane < 16) ? 0 : 2;     // A/B fragment K split per half-wave
  int oc0   = wv * 32;

  v8f acc0 = {};
  v8f acc1 = {};

  int sm = tid & 15;                   // sampler: spatial index
  int sg = tid >> 4;                   // sampler: c_local 0..15

  for (int c0 = 0; c0 < CIN; c0 += 16) {
    // prefetch next chunk's B panel slice while we sample (global_prefetch_b8)
    if (c0 + 16 < CIN)
      __builtin_prefetch(Wt + (size_t)((c0 + 16) * KKN) * COUT + tid * 16, 0, 3);

    // ---- cooperative bilinear sampling into LDS (16 c * 9 kk * 16 m) ----
    const float* xc = x + ((size_t)(b * CIN + c0 + sg)) * HW;
    #pragma unroll
    for (int kk = 0; kk < KKN; ++kk) {
      int p = kk * 16 + sm;
      float v = sW[0][p] * xc[sO[0][p]] + sW[1][p] * xc[sO[1][p]]
              + sW[2][p] * xc[sO[2][p]] + sW[3][p] * xc[sO[3][p]];
      int k = sg * KKN + kk;           // local k 0..143
      A[((k >> 1) << 5) + (sm << 1) + (k & 1)] = v;
    }
    __syncthreads();

    // ---- 144 k-values = 36 WMMA steps of K=4, two oc tiles per wave ----
    const float* Wb = Wt + (size_t)(c0 * KKN) * COUT;   // chunk base (paired)
    #pragma unroll 4
    for (int ks = 0; ks < 36; ++ks) {
      int kp = (ks << 1) + (klo >> 1);                  // pair index for this half-wave
      v2f a = *(const v2f*)&A[(kp << 5) + (mlane << 1)];
      const v2f* bp = (const v2f*)(Wb + ((size_t)kp * COUT + oc0 + mlane) * 2);
      v2f b0 = bp[0];
      v2f b1 = bp[16];
      acc0 = __builtin_amdgcn_wmma_f32_16x16x4_f32(
          false, a, false, b0, (short)0, acc0, false, false);
      acc1 = __builtin_amdgcn_wmma_f32_16x16x4_f32(
          false, a, false, b1, (short)0, acc1, false, false);
    }
    __syncthreads();
  }

  // ---- epilogue: D layout (VGPR r -> M = r + 8*hi, N = lane%16) + bias ----
  int ocA = oc0 + mlane;
  int ocB = oc0 + 16 + mlane;
  float biasA = bias[ocA];
  float biasB = bias[ocB];
  int mbase = (lane < 16) ? 0 : 8;
  float* outb = out + (size_t)b * COUT * HW + h * WW + w0;
  #pragma unroll
  for (int r = 0; r < 8; ++r) {
    int m = mbase + r;
    outb[(size_t)ocA * HW + m] = acc0[r] + biasA;
    outb[(size_t)ocB * HW + m] = acc1[r] + biasB;
  }
}

// ---------------------------------------------------------------------------
extern "C" void kernel_launch(void* const* d_in, const int* in_sizes, int n_in,
                              void* d_out, int out_size, void* d_ws, size_t ws_size,
                              hipStream_t stream) {
  (void)in_sizes; (void)n_in; (void)out_size; (void)ws_size;
  const float* x     = (const float*)d_in[0];
  const float* w_off = (const float*)d_in[1];
  const float* b_off = (const float*)d_in[2];
  const float* w     = (const float*)d_in[3];
  const float* bias  = (const float*)d_in[4];
  float* out = (float*)d_out;

  // workspace: offsets (294912 f32) then repacked weights (589824 f32) ~3.5 MB
  float* offs = (float*)d_ws;
  float* Wt   = offs + (BN * NOFF * HW);

  offset_conv_kernel<<<(BN * HW) / 256, 256, 0, stream>>>(x, w_off, b_off, offs);
  repack_w_kernel<<<(KDIM * COUT) / 256, 256, 0, stream>>>(w, Wt);
  deform_wmma_kernel<<<BN * HH * (WW / 16), 256, 0, stream>>>(x, offs, Wt, bias, out);
}